// RGCN_LinkPredictor_16990890623386
// MI455X (gfx1250) — compile-verified
//
#include <hip/hip_runtime.h>
#include <cstdint>

// RGCN link predictor for MI455X (gfx1250, wave32).
//
// Mean-aggregation commutes with the per-relation linear transform, so we
// scatter RAW features into per-(relation,dst) sums (bandwidth-bound; the
// 205MB agg buffer is nearly L2-resident), then run a dense
// [N x 576] @ [576 x O] contraction per layer with V_WMMA_F32_16X16X4_F32
// (native fp32 WMMA), folding 1/deg, bias and ReLU into frag load/store.
// Weights are pre-swizzled into WMMA B-fragment order so the GEMM's B loads
// are single coalesced b64 loads that live in the WGP$.

typedef __attribute__((ext_vector_type(2))) float v2f;
typedef __attribute__((ext_vector_type(8))) float v8f;

#define FCH 64  // per-chunk K (F_IN == F_HID == 64); 9 chunks: 8 relations + root

__global__ void count_edges_kernel(const int* __restrict__ dst,
                                   const int* __restrict__ etype,
                                   int* __restrict__ cnt, int N, int E) {
  int e = blockIdx.x * blockDim.x + threadIdx.x;
  if (e < E) atomicAdd(&cnt[(size_t)etype[e] * N + dst[e]], 1);
}

// One wave32 per edge; 2 floats per lane -> two coalesced 128B atomic bursts.
__global__ void scatter_feat_kernel(const float* __restrict__ feat,
                                    const int* __restrict__ src,
                                    const int* __restrict__ dst,
                                    const int* __restrict__ etype,
                                    float* __restrict__ agg, int N, int E) {
  int t = blockIdx.x * blockDim.x + threadIdx.x;
  int e = t >> 5, lane = t & 31;
  if (e >= E) return;
  int s = src[e], d = dst[e], r = etype[e];
  const float* fp = feat + (size_t)s * FCH;
  float* ap = agg + ((size_t)r * N + d) * FCH;
  atomicAdd(ap + lane, fp[lane]);
  atomicAdd(ap + lane + 32, fp[lane + 32]);
}

// Pre-swizzle weights into B-fragment order:
//   Bpack[((r*16 + s)*CT + ct)*32 + lane] = { B[k][n], B[k+1][n] }
// with k = 4*s + (lane<16 ? 0 : 2), n = ct*16 + (lane&15)
// (B frag layout: lanes 0-15 hold K=k,k+1 in v0,v1; lanes 16-31 hold K=k+2,k+3,
//  matching the documented B-matrix lane-split patterns in ISA 7.12.2/7.12.4).
__global__ void pack_b_kernel(const float* __restrict__ W,     // [8,64,O]
                              const float* __restrict__ root,  // [64,O]
                              v2f* __restrict__ bpack, int O) {
  int t = blockIdx.x * blockDim.x + threadIdx.x;
  int CT = O >> 4;
  int total = 9 * 16 * CT * 32;
  if (t >= total) return;
  int lane = t & 31;
  int g = t >> 5;               // (r*16 + s)*CT + ct
  int ct = g % CT;
  int s = (g / CT) % 16;
  int r = g / (CT * 16);
  int k = 4 * s + ((lane >> 4) << 1);
  int n = ct * 16 + (lane & 15);
  const float* B = (r < 8) ? (W + (size_t)r * FCH * O) : root;
  v2f v;
  v.x = B[(size_t)k * O + n];
  v.y = B[(size_t)(k + 1) * O + n];
  bpack[t] = v;
}

// out[N,O] = sum_r (agg_r / max(cnt_r,1)) @ W_r + x @ root + bias (opt ReLU).
// One wave per 16-row group; CT = O/16 accumulator tiles kept in registers so
// each A fragment is loaded once and fed to CT back-to-back WMMAs.
template <int CT>
__global__ void rgcn_gemm_kernel(const float* __restrict__ agg,    // [8,N,64]
                                 const int* __restrict__ cnt,      // [8,N]
                                 const float* __restrict__ xin,    // [N,64]
                                 const v2f* __restrict__ bpack,    // [9,16,CT,32]
                                 const float* __restrict__ bias,   // [O]
                                 float* __restrict__ out,          // [N,O]
                                 int N, int relu) {
  constexpr int O = CT * 16;
  int gw = (blockIdx.x * blockDim.x + threadIdx.x) >> 5;
  int ntiles = N >> 4;
  if (gw >= ntiles) return;  // wave-uniform exit: EXEC all-1s for WMMA
  int lane = threadIdx.x & 31;
  int m0 = gw << 4;
  int mrow = m0 + (lane & 15);
  int khalf = (lane >> 4) << 1;  // lanes 0-15 hold K=k,k+1; 16-31 hold K=k+2,k+3

  v8f acc[CT];
#pragma unroll
  for (int ct = 0; ct < CT; ++ct) acc[ct] = (v8f){};

  for (int r = 0; r < 9; ++r) {
    float inv;
    const float* Arow;
    if (r < 8) {
      int c = cnt[(size_t)r * N + mrow];
      inv = 1.0f / (float)(c > 1 ? c : 1);
      Arow = agg + ((size_t)r * N + mrow) * FCH;
    } else {
      inv = 1.0f;
      Arow = xin + (size_t)mrow * FCH;
    }
    const v2f* bp = bpack + ((size_t)r * 16 * CT) * 32 + lane;
#pragma unroll
    for (int s = 0; s < 16; ++s) {
      v2f a = *(const v2f*)(Arow + 4 * s + khalf);
      a *= inv;
#pragma unroll
      for (int ct = 0; ct < CT; ++ct) {
        v2f b = bp[(s * CT + ct) * 32];
        acc[ct] = __builtin_amdgcn_wmma_f32_16x16x4_f32(false, a, false, b,
                                                        (short)0, acc[ct],
                                                        false, false);
      }
    }
  }

#pragma unroll
  for (int ct = 0; ct < CT; ++ct) {
    int nc = ct * 16 + (lane & 15);
    float bv = bias[nc];
#pragma unroll
    for (int v = 0; v < 8; ++v) {
      // C/D layout: VGPR v -> row m0+v (lanes 0-15) or m0+v+8 (lanes 16-31)
      int row = m0 + v + ((lane >> 4) << 3);
      float val = acc[ct][v] + bv;
      if (relu) val = fmaxf(val, 0.0f);
      out[(size_t)row * O + nc] = val;
    }
  }
}

// One wave per edge; F_OUT = 32 = one float per lane; butterfly reduce.
__global__ void decode_kernel(const float* __restrict__ z,
                              const int* __restrict__ src,
                              const int* __restrict__ dst,
                              float* __restrict__ out, int E) {
  int t = blockIdx.x * blockDim.x + threadIdx.x;
  int e = t >> 5, lane = t & 31;
  if (e >= E) return;
  float p = z[(size_t)src[e] * 32 + lane] * z[(size_t)dst[e] * 32 + lane];
#pragma unroll
  for (int off = 16; off > 0; off >>= 1) p += __shfl_xor(p, off, 32);
  if (lane == 0) out[e] = p;
}

extern "C" void kernel_launch(void* const* d_in, const int* in_sizes, int n_in,
                              void* d_out, int out_size, void* d_ws, size_t ws_size,
                              hipStream_t stream) {
  // setup_inputs() order: x, W1, root1, b1, W2, root2, b2, edge_index, edge_type
  const float* x     = (const float*)d_in[0];
  const float* W1    = (const float*)d_in[1];
  const float* root1 = (const float*)d_in[2];
  const float* b1    = (const float*)d_in[3];
  const float* W2    = (const float*)d_in[4];
  const float* root2 = (const float*)d_in[5];
  const float* b2    = (const float*)d_in[6];
  const int*   ei    = (const int*)d_in[7];   // [2, E] row-major
  const int*   et    = (const int*)d_in[8];   // [E]
  const int N = in_sizes[0] / 64;
  const int E = in_sizes[8];
  const int* src = ei;
  const int* dst = ei + E;
  float* outf = (float*)d_out;

  // Workspace layout (~246 MB):
  //   agg   : 8*N*64 f32 (reused by both layers)
  //   cnt   : 8*N   i32
  //   h     : N*64  f32
  //   z     : N*32  f32
  //   bpack : 9*16*4*32 v2f (147 KB; reused by both layers)
  float* agg = (float*)d_ws;
  int*   cnt = (int*)(agg + (size_t)8 * N * 64);
  float* h   = (float*)(cnt + (size_t)8 * N);
  float* z   = h + (size_t)N * 64;
  v2f* bpack = (v2f*)(z + (size_t)N * 32);
  const size_t aggBytes = (size_t)8 * N * 64 * sizeof(float);

  const int tpb = 256;
  const int eblocks = (int)(((long long)E * 32 + tpb - 1) / tpb);
  const int gblocks = (int)(((long long)(N / 16) * 32 + tpb - 1) / tpb);

  // ---- degree counts (shared by both layers) ----
  hipMemsetAsync(cnt, 0, (size_t)8 * N * sizeof(int), stream);
  count_edges_kernel<<<(E + tpb - 1) / tpb, tpb, 0, stream>>>(dst, et, cnt, N, E);

  // ---- layer 1: scatter x -> agg, GEMM (+ReLU) -> h ----
  hipMemsetAsync(agg, 0, aggBytes, stream);
  scatter_feat_kernel<<<eblocks, tpb, 0, stream>>>(x, src, dst, et, agg, N, E);
  pack_b_kernel<<<(9 * 16 * 4 * 32 + tpb - 1) / tpb, tpb, 0, stream>>>(W1, root1,
                                                                       bpack, 64);
  rgcn_gemm_kernel<4><<<gblocks, tpb, 0, stream>>>(agg, cnt, x, bpack, b1, h, N, 1);

  // ---- layer 2: scatter h -> agg, GEMM -> z ----
  hipMemsetAsync(agg, 0, aggBytes, stream);
  scatter_feat_kernel<<<eblocks, tpb, 0, stream>>>(h, src, dst, et, agg, N, E);
  pack_b_kernel<<<(9 * 16 * 2 * 32 + tpb - 1) / tpb, tpb, 0, stream>>>(W2, root2,
                                                                       bpack, 32);
  rgcn_gemm_kernel<2><<<gblocks, tpb, 0, stream>>>(agg, cnt, h, bpack, b2, z, N, 0);

  // ---- decode: per-edge dot product ----
  decode_kernel<<<eblocks, tpb, 0, stream>>>(z, src, dst, outf, E);
}